// DenoisingConvNN_2D_Spatial_K_N_Location_25039659335751
// MI455X (gfx1250) — compile-verified
//
#include <hip/hip_runtime.h>
#include <cstddef>

typedef float v2f __attribute__((ext_vector_type(2)));
typedef float v8f __attribute__((ext_vector_type(8)));

#define T_TOK 4096   // 64x64 tokens after r=2 unshuffle of 128x128
#define S_SAMP 64    // 8x8 sample grid
#define KNN 9
#define NBATCH 16
#define FLT_BIG 3.4e38f

// ---------------------------------------------------------------------------
// Build F1 token features from x: F[b][c*4+dy*2+dx][t] = x[b][c][2*th+dy][2*tw+dx]
// ---------------------------------------------------------------------------
__global__ __launch_bounds__(256) void k_build_f1(const float* __restrict__ x,
                                                  float* __restrict__ F1) {
  int i = blockIdx.x * 256 + threadIdx.x;          // b*4096 + t
  if (i >= NBATCH * T_TOK) return;
  int b = i >> 12, t = i & 4095;
  int th = t >> 6, tw = t & 63;
  const float* xb = x + (size_t)b * 3 * 128 * 128;
  float* Fb = F1 + (size_t)b * 20 * T_TOK;
#pragma unroll
  for (int c = 0; c < 3; ++c)
#pragma unroll
    for (int dy = 0; dy < 2; ++dy)
#pragma unroll
      for (int dx = 0; dx < 2; ++dx)
        Fb[(size_t)(c * 4 + dy * 2 + dx) * T_TOK + t] =
            xb[(size_t)c * 16384 + (th * 2 + dy) * 128 + (tw * 2 + dx)];
}

// ---------------------------------------------------------------------------
// Fill the 8 coordinate channels (gx then gy, each unshuffled by r=2).
// ---------------------------------------------------------------------------
__global__ __launch_bounds__(256) void k_coords(float* __restrict__ F, int C2, int coff) {
  int i = blockIdx.x * 256 + threadIdx.x;
  if (i >= NBATCH * T_TOK) return;
  int b = i >> 12, t = i & 4095;
  int th = t >> 6, tw = t & 63;
  float* Fb = F + (size_t)b * C2 * T_TOK;
#pragma unroll
  for (int dy = 0; dy < 2; ++dy)
#pragma unroll
    for (int dx = 0; dx < 2; ++dx) {
      Fb[(size_t)(coff + 0 + dy * 2 + dx) * T_TOK + t] = (float)(tw * 2 + dx) * (1.0f / 127.0f);
      Fb[(size_t)(coff + 4 + dy * 2 + dx) * T_TOK + t] = (float)(th * 2 + dy) * (1.0f / 127.0f);
    }
}

// ---------------------------------------------------------------------------
// Gather samples at the 8x8 grid (positions 9*sy, 9*sx) + half squared norms.
// One block per batch image.
// ---------------------------------------------------------------------------
__global__ __launch_bounds__(256) void k_samp(const float* __restrict__ F, int C2,
                                              float* __restrict__ samp,
                                              float* __restrict__ hn) {
  __shared__ float s_smp[136 * S_SAMP];
  int b = blockIdx.x;
  const float* Fb = F + (size_t)b * C2 * T_TOK;
  float* sb = samp + (size_t)b * C2 * S_SAMP;
  for (int i = threadIdx.x; i < C2 * S_SAMP; i += blockDim.x) {
    int c2 = i >> 6, s = i & 63;
    int sy = s >> 3, sx = s & 7;
    int ts = (9 * sy) * 64 + 9 * sx;               // round(linspace(0,63,8)) = 9*i exactly
    float v = Fb[(size_t)c2 * T_TOK + ts];
    s_smp[i] = v;
    sb[i] = v;
  }
  __syncthreads();
  if (threadIdx.x < S_SAMP) {
    int s = threadIdx.x;
    float acc = 0.f;
    for (int c2 = 0; c2 < C2; ++c2) {
      float v = s_smp[c2 * S_SAMP + s];
      acc = fmaf(v, v, acc);
    }
    hn[b * S_SAMP + s] = 0.5f * acc;
  }
}

// ---------------------------------------------------------------------------
// P[b][k][s][o] = sum_c2 W[o][c2][k] * samp[b][c2][s]   (tiny per-image GEMM)
// grid = B*O blocks, 64 threads (one per sample s).
// ---------------------------------------------------------------------------
__global__ __launch_bounds__(64) void k_projP(const float* __restrict__ samp,
                                              const float* __restrict__ W,
                                              float* __restrict__ P, int C2, int O) {
  int bo = blockIdx.x;
  int b = bo / O, o = bo - b * O;
  int s = threadIdx.x;
  const float* sb = samp + (size_t)b * C2 * S_SAMP;
  float acc[KNN];
#pragma unroll
  for (int k = 0; k < KNN; ++k) acc[k] = 0.f;
  for (int c2 = 0; c2 < C2; ++c2) {
    float f = sb[c2 * S_SAMP + s];
    const float* wr = W + ((size_t)o * C2 + c2) * KNN;
#pragma unroll
    for (int k = 0; k < KNN; ++k) acc[k] = fmaf(wr[k], f, acc[k]);
  }
#pragma unroll
  for (int k = 0; k < KNN; ++k)
    P[((size_t)(b * KNN + k) * S_SAMP + s) * O + o] = acc[k];
}

// ---------------------------------------------------------------------------
// Fused: WMMA distance GEMM (16 tokens x 64 samples per wave) -> 9-NN select
// -> gather-accumulate through P -> relu/write token-space feature (or final
// pixel-shuffled output). Block = 256 threads = 8 wave32s = 128 tokens.
// ---------------------------------------------------------------------------
__global__ __launch_bounds__(256) void k_dist_topk_conv(
    const float* __restrict__ F, const float* __restrict__ samp,
    const float* __restrict__ hn, const float* __restrict__ P,
    const float* __restrict__ bias, float* __restrict__ dst,
    int C2, int O, int dstC2, int final_) {
  __shared__ float s_sc[128 * 65];   // 65-stride: kills 128-way bank conflict
  __shared__ float s_hn[S_SAMP];
  __shared__ int   s_idx[128 * KNN];

  const int b = blockIdx.y;
  const int tile = blockIdx.x;       // 32 tiles of 128 tokens
  const int tid = threadIdx.x;

  const float* Fb = F + (size_t)b * C2 * T_TOK;
  const float* sb = samp + (size_t)b * C2 * S_SAMP;
  if (tid < S_SAMP) s_hn[tid] = hn[b * S_SAMP + tid];
  __syncthreads();

  const int lane = tid & 31;
  const int wv = tid >> 5;           // wave 0..7
  const int m = lane & 15;           // M (A) / N (B) index
  const int kh = lane >> 4;          // lane-half selects K pair / M-half of C
  const int tokBase = tile * 128 + wv * 16;

  v8f zero = {0.f, 0.f, 0.f, 0.f, 0.f, 0.f, 0.f, 0.f};
  v8f acc[4];
#pragma unroll
  for (int j = 0; j < 4; ++j) acc[j] = zero;

  // D[m][n] += sum_{k=0..3} A[m][kk+k] * B[kk+k][n], A = F^T tile, B = samp
  for (int kk = 0; kk < C2; kk += 4) {
    const int c0 = kk + 2 * kh;
    v2f a;
    a.x = Fb[(size_t)c0 * T_TOK + tokBase + m];
    a.y = Fb[(size_t)(c0 + 1) * T_TOK + tokBase + m];
#pragma unroll
    for (int j = 0; j < 4; ++j) {
      v2f bb;
      bb.x = sb[c0 * S_SAMP + j * 16 + m];
      bb.y = sb[(c0 + 1) * S_SAMP + j * 16 + m];
      acc[j] = __builtin_amdgcn_wmma_f32_16x16x4_f32(
          /*neg_a=*/false, a, /*neg_b=*/false, bb,
          /*c_mod=*/(short)0, acc[j], /*reuse_a=*/false, /*reuse_b=*/false);
    }
  }

  // score[t][s] = 0.5*|samp_s|^2 - dot(f_t, samp_s)   (same ordering as d)
#pragma unroll
  for (int j = 0; j < 4; ++j) {
    const int s0 = j * 16 + m;
    const float h = s_hn[s0];
#pragma unroll
    for (int v = 0; v < 8; ++v) {
      const int tok = wv * 16 + kh * 8 + v;        // C/D layout: M = v + 8*laneHalf
      s_sc[tok * 65 + s0] = h - acc[j][v];
    }
  }
  __syncthreads();

  // 9-NN selection, stable (first-occurrence min == top_k tie-break)
  if (tid < 128) {
    float* row = &s_sc[tid * 65];
#pragma unroll
    for (int k = 0; k < KNN; ++k) {
      float best = FLT_BIG;
      int bi = 0;
      for (int s = 0; s < S_SAMP; ++s) {
        float v = row[s];
        if (v < best) { best = v; bi = s; }
      }
      s_idx[tid * KNN + k] = bi;
      row[bi] = FLT_BIG;
    }
  }
  __syncthreads();

  // out[o][t] = bias[o] + sum_k P[k][idx_k][o]
  const float* Pb = P + (size_t)b * KNN * S_SAMP * O;
  const int total = 128 * O;
  for (int p = tid; p < total; p += 256) {
    const int tok = p / O;
    const int o = p - tok * O;
    float a = bias[o];
#pragma unroll
    for (int k = 0; k < KNN; ++k) {
      const int s = s_idx[tok * KNN + k];
      a += Pb[((size_t)k * S_SAMP + s) * O + o];
    }
    const int t = tile * 128 + tok;
    if (final_) {
      // pixel_shuffle r=2 straight into the [16,3,128,128] output
      const int oc = o >> 2, dy = (o >> 1) & 1, dx = o & 1;
      const int th = t >> 6, tw = t & 63;
      dst[(((size_t)b * 3 + oc) * 128 + (th * 2 + dy)) * 128 + (tw * 2 + dx)] = a;
    } else {
      a = a > 0.f ? a : 0.f;                        // relu
      dst[((size_t)b * dstC2 + o) * T_TOK + t] = a; // token-space feature
    }
  }
}

// ---------------------------------------------------------------------------
extern "C" void kernel_launch(void* const* d_in, const int* in_sizes, int n_in,
                              void* d_out, int out_size, void* d_ws, size_t ws_size,
                              hipStream_t stream) {
  const float* x  = (const float*)d_in[0];
  const float* W1 = (const float*)d_in[1];
  const float* b1 = (const float*)d_in[2];
  const float* W2 = (const float*)d_in[3];
  const float* b2 = (const float*)d_in[4];
  const float* W3 = (const float*)d_in[5];
  const float* b3 = (const float*)d_in[6];
  float* out = (float*)d_out;

  float* ws = (float*)d_ws;
  float* F1   = ws;                                   // 16*20*4096
  float* F2   = F1 + (size_t)NBATCH * 20 * T_TOK;     // 16*72*4096
  float* F3   = F2 + (size_t)NBATCH * 72 * T_TOK;     // 16*136*4096
  float* samp = F3 + (size_t)NBATCH * 136 * T_TOK;    // 16*136*64
  float* hn   = samp + (size_t)NBATCH * 136 * S_SAMP; // 16*64
  float* P    = hn + (size_t)NBATCH * S_SAMP;         // 16*9*64*128

  const int nTok = NBATCH * T_TOK;
  const int nBlk = (nTok + 255) / 256;

  k_build_f1<<<nBlk, 256, 0, stream>>>(x, F1);
  k_coords<<<nBlk, 256, 0, stream>>>(F1, 20, 12);
  k_coords<<<nBlk, 256, 0, stream>>>(F2, 72, 64);
  k_coords<<<nBlk, 256, 0, stream>>>(F3, 136, 128);

  // layer 1: C2=20, O=64 -> F2
  k_samp<<<NBATCH, 256, 0, stream>>>(F1, 20, samp, hn);
  k_projP<<<NBATCH * 64, 64, 0, stream>>>(samp, W1, P, 20, 64);
  k_dist_topk_conv<<<dim3(32, NBATCH), 256, 0, stream>>>(F1, samp, hn, P, b1, F2,
                                                         20, 64, 72, 0);
  // layer 2: C2=72, O=128 -> F3
  k_samp<<<NBATCH, 256, 0, stream>>>(F2, 72, samp, hn);
  k_projP<<<NBATCH * 128, 64, 0, stream>>>(samp, W2, P, 72, 128);
  k_dist_topk_conv<<<dim3(32, NBATCH), 256, 0, stream>>>(F2, samp, hn, P, b2, F3,
                                                         72, 128, 136, 0);
  // layer 3: C2=136, O=12 -> final shuffled output
  k_samp<<<NBATCH, 256, 0, stream>>>(F3, 136, samp, hn);
  k_projP<<<NBATCH * 12, 64, 0, stream>>>(samp, W3, P, 136, 12);
  k_dist_topk_conv<<<dim3(32, NBATCH), 256, 0, stream>>>(F3, samp, hn, P, b3, out,
                                                         136, 12, 0, 1);
}